// RMILoss_9594956939903
// MI455X (gfx1250) — compile-verified
//
#include <hip/hip_runtime.h>
#include <hip/hip_bf16.h>
#include <math.h>

typedef __attribute__((ext_vector_type(2))) float v2f;
typedef __attribute__((ext_vector_type(8))) float v8f;

#define NCLS   4
#define NBAT   2
#define NC     8                  // n*c pairs
#define SD     96                 // source spatial dim
#define PD     48                 // pooled dim
#define PVOL   (PD*PD*PD)         // 110592
#define VD     46                 // valid window dim (48 - 3 + 1)
#define MTOT   (VD*VD*VD)         // 97336
#define HD     27                 // 3^3 shifts
#define ALPHA  0.0005
#define CLIP   1e-6f
#define ACC_PER_NC 3136           // Sll[1024] Spp[1024] Slp[1024] sl[32] sp[32] (doubles)
#define NBLK   8                  // chunks per (n,c): 64 blocks total
#define NWAVES 8                  // waves per block

static __device__ __forceinline__ v8f wmma4(v2f a, v2f b, v8f c) {
    // D(16x16,f32) = A(16x4,f32) x B(4x16,f32) + C
    return __builtin_amdgcn_wmma_f32_16x16x4_f32(
        /*neg_a=*/false, a, /*neg_b=*/false, b,
        /*c_mod=*/(short)0, c, /*reuse_a=*/false, /*reuse_b=*/false);
}

// ---------------------------------------------------------------- init ----
__global__ void k_init(double* acc, float* zpad, float* out) {
    int t = blockIdx.x * blockDim.x + threadIdx.x;
    int n = NC * ACC_PER_NC;
    for (int i = t; i < n; i += gridDim.x * blockDim.x) acc[i] = 0.0;
    if (t < 64) zpad[t] = 0.0f;
    if (t == 0) out[0] = 0.0f;
}

// ------------------------------------------- fused onehot/sigmoid/pool ----
__global__ void k_pool(const float* __restrict__ logits,
                       const long long* __restrict__ labels,
                       float* __restrict__ la, float* __restrict__ pr) {
    int t = blockIdx.x * blockDim.x + threadIdx.x;
    if (t >= NC * PVOL) return;
    int z = t % PD;
    int w = (t / PD) % PD;
    int h = (t / (PD * PD)) % PD;
    int c = (t / PVOL) % NCLS;
    int n = t / (PVOL * NCLS);

    float mp = 0.0f, ml = 0.0f;
#pragma unroll
    for (int dh = 0; dh < 2; ++dh)
#pragma unroll
    for (int dw = 0; dw < 2; ++dw)
#pragma unroll
    for (int dz = 0; dz < 2; ++dz) {
        int H = 2 * h + dh, W = 2 * w + dw, Z = 2 * z + dz;
        long long lidx = (((long long)n * SD + H) * SD + W) * SD + Z;
        long long gidx = ((((long long)n * NCLS + c) * SD + H) * SD + W) * SD + Z;
        int lab = (int)labels[lidx];
        float x = logits[gidx];
        float s = 1.0f / (1.0f + __expf(-x));
        mp = fmaxf(mp, s);
        if (lab == c) ml = 1.0f;
    }
    la[t] = ml;
    pr[t] = mp + CLIP;
}

// ------------------------------------------------ WMMA Gram matrices ----
// Per (n,c): S_ll, S_pp (symmetric, upper tiles), S_lp (full 2x2 tiles) over
// the 27(->32)-row pair-vector matrix, plus row sums for mean-centering.
__global__ void k_gram(const float* __restrict__ laAll,
                       const float* __restrict__ prAll,
                       const float* __restrict__ zpad,
                       double* __restrict__ accAll) {
    const int nc = blockIdx.x & 7;
    const int chunk = blockIdx.x >> 3;
    const float* la = laAll + nc * PVOL;
    const float* pr = prAll + nc * PVOL;
    double* acc = accAll + nc * ACC_PER_NC;

    const int wv = threadIdx.x >> 5;
    const int lane = threadIdx.x & 31;
    const int wg = chunk * NWAVES + wv;          // 0..63 global wave id per nc

    const int r    = lane & 15;
    const int hs   = lane >> 4;
    const int koff = hs * 2;                     // K offset handled by this half-wave
    const int d0 = r, d1 = 16 + r;
    const int i0 = d0 / 9, j0 = (d0 / 3) % 3, k0 = d0 % 3;
    const bool v1 = (d1 < HD);
    const int dd1 = v1 ? d1 : 0;
    const int i1 = dd1 / 9, j1 = (dd1 / 3) % 3, k1 = dd1 % 3;
    const bool m0 = (koff == 0);                 // tail z-step validity

    v8f cll00 = {}, cll01 = {}, cll11 = {};
    v8f cpp00 = {}, cpp01 = {}, cpp11 = {};
    v8f clp00 = {}, clp01 = {}, clp10 = {}, clp11 = {};
    float sla0 = 0.f, sla1 = 0.f, spr0 = 0.f, spr1 = 0.f;

    // Walk (h,w) rows of the 46x46x46 window; per row, z advances 4 per step
    // (lane's two K slots = z + koff + {0,1}); 11 full steps + 1 masked tail.
    for (int q = wg; q < VD * VD; q += NC * NWAVES) {
        int h = q / VD;
        int w = q - h * VD;
        int base0 = ((h + i0) * PD + (w + j0)) * PD + k0 + koff;
        int base1 = ((h + i1) * PD + (w + j1)) * PD + k1 + koff;
        const float* pla0 = la + base0;
        const float* ppr0 = pr + base0;
        const float* pla1 = v1 ? (la + base1) : zpad;
        const float* ppr1 = v1 ? (pr + base1) : zpad;

#pragma unroll 1
        for (int s = 0; s < 11; ++s) {
            v2f a0, a1, p0, p1;
            a0.x = pla0[0]; a0.y = pla0[1];
            a1.x = pla1[0]; a1.y = pla1[1];
            p0.x = ppr0[0]; p0.y = ppr0[1];
            p1.x = ppr1[0]; p1.y = ppr1[1];
            pla0 += 4; pla1 += 4; ppr0 += 4; ppr1 += 4;

            sla0 += a0.x + a0.y; sla1 += a1.x + a1.y;
            spr0 += p0.x + p0.y; spr1 += p1.x + p1.y;

            cll00 = wmma4(a0, a0, cll00);
            cll01 = wmma4(a0, a1, cll01);
            cll11 = wmma4(a1, a1, cll11);
            cpp00 = wmma4(p0, p0, cpp00);
            cpp01 = wmma4(p0, p1, cpp01);
            cpp11 = wmma4(p1, p1, cpp11);
            clp00 = wmma4(a0, p0, clp00);
            clp01 = wmma4(a0, p1, clp01);
            clp10 = wmma4(a1, p0, clp10);
            clp11 = wmma4(a1, p1, clp11);
        }

        // tail: z = 44 + koff + {0,1}; valid only for koff==0 lanes
        {
            v2f a0, a1, p0, p1;
            a0.x = m0 ? pla0[0] : 0.f; a0.y = m0 ? pla0[1] : 0.f;
            a1.x = m0 ? pla1[0] : 0.f; a1.y = m0 ? pla1[1] : 0.f;
            p0.x = m0 ? ppr0[0] : 0.f; p0.y = m0 ? ppr0[1] : 0.f;
            p1.x = m0 ? ppr1[0] : 0.f; p1.y = m0 ? ppr1[1] : 0.f;

            sla0 += a0.x + a0.y; sla1 += a1.x + a1.y;
            spr0 += p0.x + p0.y; spr1 += p1.x + p1.y;

            cll00 = wmma4(a0, a0, cll00);
            cll01 = wmma4(a0, a1, cll01);
            cll11 = wmma4(a1, a1, cll11);
            cpp00 = wmma4(p0, p0, cpp00);
            cpp01 = wmma4(p0, p1, cpp01);
            cpp11 = wmma4(p1, p1, cpp11);
            clp00 = wmma4(a0, p0, clp00);
            clp01 = wmma4(a0, p1, clp01);
            clp10 = wmma4(a1, p0, clp10);
            clp11 = wmma4(a1, p1, clp11);
        }
    }

    __shared__ float red[10 * 256];
    __shared__ float sred[64];
    for (int i = threadIdx.x; i < 2560; i += 256) red[i] = 0.f;
    if (threadIdx.x < 64) sred[threadIdx.x] = 0.f;
    __syncthreads();

    const int rowbase = hs * 8;
#define EMIT(T, idx)                                                         \
    _Pragma("unroll") for (int v = 0; v < 8; ++v)                            \
        atomicAdd(&red[(idx) * 256 + (rowbase + v) * 16 + r], (T)[v]);
    EMIT(cll00, 0) EMIT(cll01, 1) EMIT(cll11, 2)
    EMIT(cpp00, 3) EMIT(cpp01, 4) EMIT(cpp11, 5)
    EMIT(clp00, 6) EMIT(clp01, 7) EMIT(clp10, 8) EMIT(clp11, 9)
#undef EMIT
    atomicAdd(&sred[d0], sla0);      atomicAdd(&sred[d1], sla1);
    atomicAdd(&sred[32 + d0], spr0); atomicAdd(&sred[32 + d1], spr1);
    __syncthreads();

    const int mbase[10] = {0, 0, 0, 1024, 1024, 1024, 2048, 2048, 2048, 2048};
    const int rb[10]    = {0, 0, 16, 0, 0, 16, 0, 0, 16, 16};
    const int cb[10]    = {0, 16, 16, 0, 16, 16, 0, 16, 0, 16};
    for (int i = threadIdx.x; i < 2560; i += 256) {
        int t = i >> 8; int e = i & 255; int rr = e >> 4; int cc = e & 15;
        atomicAdd(&acc[mbase[t] + (rb[t] + rr) * 32 + (cb[t] + cc)], (double)red[i]);
    }
    if (threadIdx.x < 64)
        atomicAdd(&acc[3072 + threadIdx.x], (double)sred[threadIdx.x]);
}

// -------------------------- f64 finalize: inv, Schur, Cholesky, logdet ----
__global__ void k_fin(const double* __restrict__ accAll, float* __restrict__ out) {
    const int nc = blockIdx.x;
    const double* acc = accAll + nc * ACC_PER_NC;
    __shared__ double A[HD * HD], P[HD * HD], G[HD * HD], IN[HD * HD], T[HD * HD];
    __shared__ double rs[32];
    const int lane = threadIdx.x;
    const double invM = 1.0 / (double)MTOT;

    for (int idx = lane; idx < HD * HD; idx += 32) {
        int d = idx / HD, e = idx % HD;
        double sld = acc[3072 + d], sle = acc[3072 + e];
        double spd = acc[3104 + d], spe = acc[3104 + e];
        double ll = (d >= 16 && e < 16) ? acc[e * 32 + d]        : acc[d * 32 + e];
        double pp = (d >= 16 && e < 16) ? acc[1024 + e * 32 + d] : acc[1024 + d * 32 + e];
        double lp = acc[2048 + d * 32 + e];
        A[idx]  = ll - sld * sle * invM;
        P[idx]  = pp - spd * spe * invM + (d == e ? ALPHA : 0.0);
        G[idx]  = lp - sld * spe * invM;
        IN[idx] = (d == e) ? 1.0 : 0.0;
    }
    __syncthreads();

    // Gauss-Jordan inverse of P (SPD + alpha*I: no pivoting needed)
    for (int k = 0; k < HD; ++k) {
        double ip = 1.0 / P[k * HD + k];
        __syncthreads();
        if (lane < HD) { P[k * HD + lane] *= ip; IN[k * HD + lane] *= ip; }
        __syncthreads();
        for (int rr = 0; rr < HD; ++rr) {
            if (rr == k) continue;
            double f = P[rr * HD + k];
            if (lane < HD) {
                P[rr * HD + lane]  -= f * P[k * HD + lane];
                IN[rr * HD + lane] -= f * IN[k * HD + lane];
            }
        }
        __syncthreads();
    }

    // T = G * inv(P)
    for (int idx = lane; idx < HD * HD; idx += 32) {
        int d = idx / HD, e = idx % HD;
        double s = 0.0;
        for (int k = 0; k < HD; ++k) s += G[d * HD + k] * IN[k * HD + e];
        T[idx] = s;
    }
    __syncthreads();

    // P := A - T*G^T + alpha*I   (appro_var + alpha I)
    for (int idx = lane; idx < HD * HD; idx += 32) {
        int d = idx / HD, e = idx % HD;
        double s = 0.0;
        for (int k = 0; k < HD; ++k) s += T[d * HD + k] * G[e * HD + k];
        P[idx] = A[idx] - s + (d == e ? ALPHA : 0.0);
    }
    __syncthreads();

    // in-place right-looking Cholesky on P
    for (int k = 0; k < HD; ++k) {
        if (lane == 0) P[k * HD + k] = sqrt(P[k * HD + k]);
        __syncthreads();
        double dk = P[k * HD + k];
        if (lane > k && lane < HD) P[lane * HD + k] /= dk;
        __syncthreads();
        for (int c = k + 1; c < HD; ++c)
            if (lane >= c && lane < HD)
                P[lane * HD + c] -= P[lane * HD + k] * P[c * HD + k];
        __syncthreads();
    }

    double s = 0.0;
    if (lane < HD) s = log(P[lane * HD + lane] + 1e-8);
    rs[lane] = s;
    __syncthreads();
    if (lane == 0) {
        double tot = 0.0;
        for (int i = 0; i < HD; ++i) tot += rs[i];
        atomicAdd(out, (float)(tot / (double)(NBAT * HD)));
    }
}

// ----------------------------------------------------------------------
extern "C" void kernel_launch(void* const* d_in, const int* in_sizes, int n_in,
                              void* d_out, int out_size, void* d_ws, size_t ws_size,
                              hipStream_t stream) {
    const float* logits = (const float*)d_in[0];
    const long long* labels = (const long long*)d_in[1];
    float* out = (float*)d_out;

    float* laW = (float*)d_ws;
    float* prW = laW + NC * PVOL;
    double* accW = (double*)(prW + NC * PVOL);        // byte offset 7077888, 8B aligned
    float* zpadW = (float*)(accW + NC * ACC_PER_NC);  // 64 zero floats

    k_init<<<64, 256, 0, stream>>>(accW, zpadW, out);
    k_pool<<<(NC * PVOL + 255) / 256, 256, 0, stream>>>(logits, labels, laW, prW);
    k_gram<<<NC * NBLK, 256, 0, stream>>>(laW, prW, zpadW, accW);
    k_fin<<<NC, 32, 0, stream>>>(accW, out);
}